// InteractionArch_39393440039438
// MI455X (gfx1250) — compile-verified
//
#include <hip/hip_runtime.h>

typedef __attribute__((ext_vector_type(2))) float v2f;
typedef __attribute__((ext_vector_type(8))) float v8f;

#define NF     26     // sparse features
#define DD     128    // feature dim
#define NROWS  27     // 1 dense + 26 sparse
#define OUTW   479    // 128 + 27*26/2
#define LDS_STRIDE 132  // 128 + 4 pad: LDS bank stride 4 -> conflict-free 16-row b64 reads

__global__ __launch_bounds__(128) void dlrm_interact_wmma(
    const float* __restrict__ dense,
    const float* __restrict__ sparse,
    float* __restrict__ out)
{
    __shared__ float xs[32 * LDS_STRIDE];   // 16896 B of the 320 KB WGP LDS

    const int tid  = threadIdx.x;
    const int b    = blockIdx.x;
    const int lane = tid & 31;              // wave32
    const int wave = tid >> 5;

    // ---- zero-pad rows 27..31 via plain DS stores (DScnt, covered by barrier) ----
    for (int i = tid; i < 5 * 32; i += 128) {
        int row = NROWS + (i >> 5);
        int c4  = i & 31;
        *reinterpret_cast<float4*>(&xs[row * LDS_STRIDE + c4 * 4]) =
            make_float4(0.f, 0.f, 0.f, 0.f);
    }

    // ---- async global -> LDS staging for rows 0..26 (ASYNCcnt, bypasses VGPRs) ----
    const float* dbase = dense  + (size_t)b * DD;
    const float* sbase = sparse + (size_t)b * (NF * DD);
    #pragma unroll
    for (int i = 0; i < 7; ++i) {
        int idx = i * 128 + tid;                 // float4 slot, 0..895
        if (idx < NROWS * 32) {
            int row = idx >> 5;                  // 0..26
            int c4  = idx & 31;
            const float* src = (row == 0) ? (dbase + c4 * 4)
                                          : (sbase + (row - 1) * DD + c4 * 4);
            // low 32 bits of a generic shared pointer = LDS byte offset
            unsigned lds_addr =
                (unsigned)(uintptr_t)(&xs[row * LDS_STRIDE + c4 * 4]);
            asm volatile("global_load_async_to_lds_b128 %0, %1, off"
                         :: "v"(lds_addr), "v"(src)
                         : "memory");
        }
    }
    // __syncthreads waits DScnt only; async-LDS traffic needs an ASYNCcnt wait.
    asm volatile("s_wait_asynccnt 0x0" ::: "memory");
    __syncthreads();

    const size_t obase = (size_t)b * OUTW;

    if (wave == 3) {
        // dense passthrough: out[b, 0:128]
        int l = tid - 96;
        #pragma unroll
        for (int j = 0; j < 4; ++j)
            out[obase + l + j * 32] = xs[l + j * 32];
        return;
    }

    // waves 0..2 own 16x16 tile pairs (mt,nt) = (0,0), (0,1), (1,1)
    const int mt = (wave == 2) ? 1 : 0;
    const int nt = (wave == 0) ? 0 : 1;

    // ISA f32 A/B fragment layout: lane&15 selects row within tile,
    // lane>>4 selects K sub-pair {0,1} vs {2,3} of each K=4 step.
    const int r    = lane & 15;
    const int half = lane >> 4;
    const float* arow = &xs[(mt * 16 + r) * LDS_STRIDE + half * 2];
    const float* brow = &xs[(nt * 16 + r) * LDS_STRIDE + half * 2];

    v8f acc = {};
    #pragma unroll 8
    for (int kb = 0; kb < DD; kb += 4) {
        v2f a  = *reinterpret_cast<const v2f*>(arow + kb);
        v2f bb = *reinterpret_cast<const v2f*>(brow + kb);
        // D = A * B + C in f32: v_wmma_f32_16x16x4_f32
        acc = __builtin_amdgcn_wmma_f32_16x16x4_f32(
                  /*neg_a=*/false, a, /*neg_b=*/false, bb,
                  /*c_mod=*/(short)0, acc, /*reuse_a=*/false, /*reuse_b=*/false);
    }

    // ---- scatter strict-upper-triangular entries ----
    // D layout: VGPR j, lanes 0-15 -> M=j; lanes 16-31 -> M=j+8; N = lane&15.
    const int n = nt * 16 + r;
    #pragma unroll
    for (int j = 0; j < 8; ++j) {
        int m = mt * 16 + half * 8 + j;
        if (m < n && n < NROWS) {
            int t = m * 26 - (m * (m - 1)) / 2 + (n - m - 1);
            out[obase + 128 + t] = acc[j];
        }
    }
}

extern "C" void kernel_launch(void* const* d_in, const int* in_sizes, int n_in,
                              void* d_out, int out_size, void* d_ws, size_t ws_size,
                              hipStream_t stream) {
    const float* dense  = (const float*)d_in[0];   // (B, 128) f32
    const float* sparse = (const float*)d_in[1];   // (B, 26*128) f32
    float* out = (float*)d_out;                    // (B, 479) f32
    const int B = in_sizes[0] / DD;                // 16384
    dlrm_interact_wmma<<<B, 128, 0, stream>>>(dense, sparse, out);
}